// Repeat_Recommendation_Decoder_23132693856435
// MI455X (gfx1250) — compile-verified
//
#include <hip/hip_runtime.h>

#define S_DIM 50
#define H_DIM 64
#define NUM_ITEM 40000

typedef __attribute__((ext_vector_type(2))) float v2f;
typedef __attribute__((ext_vector_type(4))) float v4f;
typedef __attribute__((ext_vector_type(8))) float v8f;

// ---------------------------------------------------------------------------
// Kernel 1: stream 655MB of zeros at full HBM rate (lean: max occupancy,
// non-temporal B128 stores so the output doesn't pollute L2).
// ---------------------------------------------------------------------------
__global__ __launch_bounds__(256) void zero_out_kernel(float* __restrict__ out,
                                                       long n4) {
    const v4f z = {0.f, 0.f, 0.f, 0.f};
    v4f* p = (v4f*)out;
    long i = (long)blockIdx.x * blockDim.x + threadIdx.x;
    const long stride = (long)gridDim.x * blockDim.x;
    for (; i < n4; i += stride)
        __builtin_nontemporal_store(z, p + i);
}

// ---------------------------------------------------------------------------
// Kernel 2: one wave per batch row. WMMA f32 16x16x4 additive attention,
// softmax over S=50, scatter-add onto the (already zeroed) item row.
// ---------------------------------------------------------------------------
__global__ __launch_bounds__(32) void rr_attn_kernel(
    const float* __restrict__ all_memory,   // [B,S,H]
    const float* __restrict__ last_memory,  // [B,H]
    const float* __restrict__ Wr,           // [H,H]
    const float* __restrict__ Ur,           // [H,H]
    const float* __restrict__ Vr_w,         // [1,H]
    const float* __restrict__ Vr_b,         // [1]
    const int*   __restrict__ seq_item,     // [B,S]
    float* __restrict__ out)                // [B,NUM_ITEM]
{
    __shared__ float sm_scores[64];

    const int b    = blockIdx.x;
    const int lane = threadIdx.x;           // 0..31
    const int rL    = lane & 15;            // row/col within 16-tile
    const int kBase = (lane >> 4) << 1;     // 0 or 2: K-pair held by this half

    const float* A     = all_memory  + (size_t)b * (S_DIM * H_DIM);
    const float* lmrow = last_memory + (size_t)b * H_DIM;
    float* out_row     = out + (size_t)b * NUM_ITEM;

    // lm = last_memory[b] @ Wr^T ; lane holds lm[lane] (l0) and lm[lane+32] (l1)
    float l0 = 0.f, l1 = 0.f;
    #pragma unroll 8
    for (int h = 0; h < H_DIM; ++h) {
        const float x = lmrow[h];
        l0 = fmaf(x, Wr[ lane       * H_DIM + h], l0);
        l1 = fmaf(x, Wr[(lane + 32) * H_DIM + h], l1);
    }
    const float bias = Vr_b[0];

    for (int m = 0; m < 4; ++m) {           // tiles over S (padded 50 -> 64)
        const int  srow  = m * 16 + rL;
        const bool valid = (srow < S_DIM);
        const int  srowL = valid ? srow : (S_DIM - 1);  // clamp: branch-free load

        // Preload A tile fragments: 16 K-steps, v2f each (K = 4k+kBase, +1)
        v2f afrag[16];
        #pragma unroll
        for (int k = 0; k < 16; ++k) {
            v2f af = *(const v2f*)(A + srowL * H_DIM + k * 4 + kBase);
            afrag[k].x = valid ? af.x : 0.f;            // v_cndmask, no branch
            afrag[k].y = valid ? af.y : 0.f;
        }

        float sacc[8] = {0.f,0.f,0.f,0.f,0.f,0.f,0.f,0.f};

        #pragma unroll 1                     // keep VGPR pressure low
        for (int n = 0; n < 4; ++n) {        // tiles over output feature i
            const int icol = n * 16 + rL;
            v8f acc = {};
            #pragma unroll
            for (int k = 0; k < 16; ++k) {
                // B(kabs, icol) = Ur[icol*64 + kabs]
                v2f bfrag = *(const v2f*)(Ur + icol * H_DIM + k * 4 + kBase);
                acc = __builtin_amdgcn_wmma_f32_16x16x4_f32(
                    false, afrag[k], false, bfrag, (short)0, acc, false, false);
            }
            // broadcast lm[icol] from registers (lanes hold lm[lane], lm[lane+32])
            const float lm_i = (n < 2) ? __shfl(l0, icol, 32)
                                       : __shfl(l1, icol - 32, 32);
            const float vr_i = Vr_w[icol];
            #pragma unroll
            for (int v = 0; v < 8; ++v)
                sacc[v] += tanhf(acc[v] + lm_i) * vr_i;
        }

        // reduce over N: sum across the 16 lanes of each half-wave
        #pragma unroll
        for (int off = 8; off >= 1; off >>= 1) {
            #pragma unroll
            for (int v = 0; v < 8; ++v)
                sacc[v] += __shfl_xor(sacc[v], off, 32);
        }
        if (rL == 0) {                       // lanes 0 (M=v) and 16 (M=v+8)
            const int baseM = (lane >> 4) * 8;
            #pragma unroll
            for (int v = 0; v < 8; ++v) {
                const int s = m * 16 + baseM + v;
                if (s < S_DIM) sm_scores[s] = sacc[v] + bias;
            }
        }
    }

    __syncthreads();   // single-wave WG: hw-nop barrier, but orders LDS wr->rd

    // ---- softmax over S=50 (scores now distributed: lane -> s=lane, s=lane+32)
    float v0 = sm_scores[lane];                                    // s = lane < 50
    float v1 = (lane + 32 < S_DIM) ? sm_scores[lane + 32] : -1e30f;
    float mx = fmaxf(v0, v1);
    #pragma unroll
    for (int off = 16; off >= 1; off >>= 1)
        mx = fmaxf(mx, __shfl_xor(mx, off, 32));
    float e0 = expf(v0 - mx);
    float e1 = (lane + 32 < S_DIM) ? expf(v1 - mx) : 0.f;
    float sum = e0 + e1;
    #pragma unroll
    for (int off = 16; off >= 1; off >>= 1)
        sum += __shfl_xor(sum, off, 32);
    const float inv = 1.f / sum;

    // ---- scatter probs straight from registers (duplicates accumulate)
    const int* seq = seq_item + (size_t)b * S_DIM;
    atomicAdd(out_row + seq[lane], e0 * inv);
    if (lane + 32 < S_DIM)
        atomicAdd(out_row + seq[lane + 32], e1 * inv);
}

extern "C" void kernel_launch(void* const* d_in, const int* in_sizes, int n_in,
                              void* d_out, int out_size, void* d_ws, size_t ws_size,
                              hipStream_t stream) {
    const float* all_memory  = (const float*)d_in[0];
    const float* last_memory = (const float*)d_in[1];
    const float* Wr          = (const float*)d_in[2];
    const float* Ur          = (const float*)d_in[3];
    const float* Vr_w        = (const float*)d_in[4];
    const float* Vr_b        = (const float*)d_in[5];
    const int*   seq_item    = (const int*)d_in[6];
    float* out = (float*)d_out;

    const int  B  = in_sizes[1] / H_DIM;                 // 4096
    const long n4 = ((long)B * NUM_ITEM) / 4;            // 40,960,000 x 16B

    zero_out_kernel<<<16384, 256, 0, stream>>>(out, n4);
    rr_attn_kernel<<<B, 32, 0, stream>>>(all_memory, last_memory, Wr, Ur,
                                         Vr_w, Vr_b, seq_item, out);
}